// SpringPotential_55989193671171
// MI455X (gfx1250) — compile-verified
//
#include <hip/hip_runtime.h>

typedef __attribute__((ext_vector_type(2))) float v2f;
typedef __attribute__((ext_vector_type(8))) float v8f;

#define DIM 32
#define K_CONST 1.0f
#define L0_CONST 0.1f

__global__ void zero_kernel(float* __restrict__ p, int n) {
  int i = blockIdx.x * blockDim.x + threadIdx.x;
  if (i < n) p[i] = 0.0f;
}

__global__ __launch_bounds__(256) void spring_grad_kernel(
    const float* __restrict__ x,
    const int* __restrict__ edges,
    float* __restrict__ out,   // out[0] = energy, out+1 = grad (N*32)
    int E_total) {
  __shared__ float s_energy;
  if (threadIdx.x == 0) s_energy = 0.0f;
  __syncthreads();

  float* __restrict__ grad = out + 1;

  const int lane = threadIdx.x & 31;
  const int m    = lane & 15;   // which of the 16 edges in this chunk
  const int half = lane >> 4;   // 0: K elems {4j,4j+1}, 1: {4j+2,4j+3}
  const int k0   = 2 * half;

  const int wave   = (blockIdx.x * blockDim.x + threadIdx.x) >> 5;
  const int nwaves = (gridDim.x * blockDim.x) >> 5;
  const int nchunks = (E_total + 15) >> 4;

  float esum = 0.0f;

  for (int chunk = wave; chunk < nchunks; chunk += nwaves) {
    const int e     = chunk * 16 + m;
    const bool valid = (e < E_total);
    const int ec    = valid ? e : 0;

    const int u = edges[2 * ec + 0];
    const int v = edges[2 * ec + 1];
    const float* __restrict__ xu = x + (size_t)u * DIM;
    const float* __restrict__ xv = x + (size_t)v * DIM;

    // Lane L holds elements {4j+2*half, 4j+2*half+1} of row (L%16):
    // exactly the 16x4 fp32 A layout and (by symmetry) the 4x16 B layout.
    v2f au[8], av[8];
#pragma unroll
    for (int j = 0; j < 8; ++j) {
      const int k = 4 * j + k0;
      au[j] = *(const v2f*)(xu + k);
      av[j] = *(const v2f*)(xv + k);
    }

    // C[m][n] = sum_k xu_m[k] * J[k] * xv_n[k]; we want the diagonal.
    v8f c = {0.f, 0.f, 0.f, 0.f, 0.f, 0.f, 0.f, 0.f};
#pragma unroll
    for (int j = 0; j < 8; ++j) {
      v2f b = av[j];
      if (j == 0) b.x = half ? b.x : -b.x;  // apply J to k==0 (branchless select)
      c = __builtin_amdgcn_wmma_f32_16x16x4_f32(
          false, au[j], false, b, (short)0, c, false, false);
    }

    // Diagonal extraction: edge m's inner product lives on
    // lane m (m<8, in c[m]) or lane m+16 (m>=8, in c[m-8]).
    const int r = lane & 7;
    float sel = c[0];
    sel = (r == 1) ? c[1] : sel;
    sel = (r == 2) ? c[2] : sel;
    sel = (r == 3) ? c[3] : sel;
    sel = (r == 4) ? c[4] : sel;
    sel = (r == 5) ? c[5] : sel;
    sel = (r == 6) ? c[6] : sel;
    sel = (r == 7) ? c[7] : sel;
    const int src = (m < 8) ? m : (m + 16);
    float inner = __shfl(sel, src, 32);   // both half-lanes of edge m get it

    inner = fminf(inner, -1.0f - 1e-07f);
    const float dist   = acoshf(-inner);
    const float delta  = dist - L0_CONST;
    const float denom  = sqrtf(inner * inner - 1.0f);
    const float factor = -(K_CONST * delta / (denom + 1e-09f));

    if (valid) {
      if (half == 0) esum += 0.5f * K_CONST * delta * delta;
#pragma unroll
      for (int j = 0; j < 8; ++j) {
        const int k = 4 * j + k0;
        v2f gu = au[j];
        v2f gv = av[j];
        if (j == 0 && half == 0) { gu.x = -gu.x; gv.x = -gv.x; }  // J
        atomicAdd(&grad[(size_t)u * DIM + k],     factor * gv.x);
        atomicAdd(&grad[(size_t)u * DIM + k + 1], factor * gv.y);
        atomicAdd(&grad[(size_t)v * DIM + k],     factor * gu.x);
        atomicAdd(&grad[(size_t)v * DIM + k + 1], factor * gu.y);
      }
    }
  }

  // Wave reduction of energy, then block LDS accumulate, one global atomic.
#pragma unroll
  for (int off = 16; off > 0; off >>= 1)
    esum += __shfl_xor(esum, off, 32);
  if (lane == 0) atomicAdd(&s_energy, esum);
  __syncthreads();
  if (threadIdx.x == 0) atomicAdd(out, s_energy);
}

extern "C" void kernel_launch(void* const* d_in, const int* in_sizes, int n_in,
                              void* d_out, int out_size, void* d_ws, size_t ws_size,
                              hipStream_t stream) {
  const float* x     = (const float*)d_in[0];
  const int*   edges = (const int*)d_in[1];
  float*       out   = (float*)d_out;

  const int E_total = in_sizes[1] / 2;

  // d_out is poisoned by the harness; we accumulate with atomics, so zero it.
  zero_kernel<<<(out_size + 255) / 256, 256, 0, stream>>>(out, out_size);

  const int nchunks = (E_total + 15) / 16;           // 16 edges per wave
  const int blocks  = (nchunks + 7) / 8;             // 8 waves per 256-thr block
  spring_grad_kernel<<<blocks, 256, 0, stream>>>(x, edges, out, E_total);
}